// CanonicalAlignmentLoss_88596585382423
// MI455X (gfx1250) — compile-verified
//
#include <hip/hip_runtime.h>

// ---------------------------------------------------------------------------
// CanonicalAlignmentLoss for MI455X (gfx1250, wave32)
//   Phase 0: zero accumulators in workspace
//   Phase 1: per-subject masked Gram via V_WMMA_F32_16X16X4_F32
//            (16 waves per block, one subject per wave, tile shared via LDS)
//   Phase 2: finalize (means -> cov -> pairwise Frobenius -> mean loss)
// ---------------------------------------------------------------------------

#define NSUBJ 16
#define DIM   64
#define CHUNK 16   // rows per LDS tile

typedef __attribute__((ext_vector_type(2))) float v2f;
typedef __attribute__((ext_vector_type(8))) float v8f;

// workspace layout (floats): gram[16][64][64] | sums[16][64] | counts[16]
#define GRAM_OFF 0
#define SUMS_OFF (NSUBJ * DIM * DIM)
#define CNT_OFF  (SUMS_OFF + NSUBJ * DIM)
#define WS_FLOATS (CNT_OFF + NSUBJ)

__global__ void caloss_zero_ws(float* __restrict__ ws) {
  int i = blockIdx.x * blockDim.x + threadIdx.x;
  if (i < WS_FLOATS) ws[i] = 0.0f;
}

__global__ __launch_bounds__(512) void caloss_gram_kernel(
    const float* __restrict__ emb, const int* __restrict__ sid,
    float* __restrict__ ws, int N) {
  __shared__ __align__(16) float tile[CHUNK * DIM];  // 4 KB
  __shared__ int ssid[CHUNK];

  const int lane = threadIdx.x & 31;
  const int subj = threadIdx.x >> 5;  // wave id == subject id (16 waves)

  v8f acc[16] = {};                   // 4x4 grid of 16x16 f32 accumulators
  float sum_lo = 0.0f, sum_hi = 0.0f, cnt = 0.0f;

  const int m  = lane & 15;           // matrix row within fragment
  const int kh = (lane >> 4) << 1;    // K offset: 0 (lanes 0-15) or 2 (16-31)

  const int nchunks = N / CHUNK;
  for (int c = blockIdx.x; c < nchunks; c += gridDim.x) {
    const int row0 = c * CHUNK;
    // cooperative tile load: 1024 contiguous floats -> 512 threads x float2
    ((float2*)tile)[threadIdx.x] =
        ((const float2*)(emb + (size_t)row0 * DIM))[threadIdx.x];
    if (threadIdx.x < CHUNK) ssid[threadIdx.x] = sid[row0 + threadIdx.x];
    __syncthreads();

    // wave-uniform presence test for this subject in the chunk
    bool present = false;
#pragma unroll
    for (int r = 0; r < CHUNK; r++) present = present || (ssid[r] == subj);

    if (present) {
#pragma unroll
      for (int ks = 0; ks < 4; ks++) {        // 4 rows per WMMA (K=4)
        const int rA = ks * 4 + kh;
        const float mask0 = (ssid[rA] == subj) ? 1.0f : 0.0f;
        const float mask1 = (ssid[rA + 1] == subj) ? 1.0f : 0.0f;
        v2f fa[4], fb[4];                     // frags for the 4 d-blocks
#pragma unroll
        for (int b = 0; b < 4; b++) {
          float x0 = tile[rA * DIM + b * 16 + m];
          float x1 = tile[(rA + 1) * DIM + b * 16 + m];
          v2f vb; vb.x = x0;         vb.y = x1;
          v2f va; va.x = x0 * mask0; va.y = x1 * mask1;
          fb[b] = vb;
          fa[b] = va;  // masked side (one side suffices for the product)
        }
#pragma unroll
        for (int bi = 0; bi < 4; bi++)
#pragma unroll
          for (int bj = 0; bj < 4; bj++)
            acc[bi * 4 + bj] = __builtin_amdgcn_wmma_f32_16x16x4_f32(
                false, fa[bi], false, fb[bj], (short)0, acc[bi * 4 + bj],
                false, false);
      }
      // per-subject sums (lane covers dims lane and lane+32) and count
#pragma unroll
      for (int r = 0; r < CHUNK; r++) {
        if (ssid[r] == subj) {
          sum_lo += tile[r * DIM + lane];
          sum_hi += tile[r * DIM + lane + 32];
          cnt += 1.0f;  // identical across lanes; only lane 0 flushes it
        }
      }
    }
    __syncthreads();
  }

  // flush partial results: 16x16 C/D layout -> lanes 0-15: M=v, 16-31: M=v+8
  float* gram = ws + GRAM_OFF + subj * DIM * DIM;
  const int nn = lane & 15;
  const int mbase = (lane >> 4) * 8;
#pragma unroll
  for (int bi = 0; bi < 4; bi++)
#pragma unroll
    for (int bj = 0; bj < 4; bj++) {
      v8f a = acc[bi * 4 + bj];
#pragma unroll
      for (int v = 0; v < 8; v++) {
        int d1 = bi * 16 + mbase + v;
        int d2 = bj * 16 + nn;
        atomicAdd(&gram[d1 * DIM + d2], a[v]);
      }
    }
  float* sums = ws + SUMS_OFF + subj * DIM;
  atomicAdd(&sums[lane], sum_lo);
  atomicAdd(&sums[lane + 32], sum_hi);
  if (lane == 0) atomicAdd(&ws[CNT_OFF + subj], cnt);
}

__global__ __launch_bounds__(256) void caloss_finalize(
    const float* __restrict__ ws, float* __restrict__ out) {
  __shared__ float means[NSUBJ * DIM];
  __shared__ float nshr[NSUBJ];
  __shared__ float red[256];
  const int t = threadIdx.x;

  if (t < NSUBJ) nshr[t] = ws[CNT_OFF + t];
  __syncthreads();
  for (int i = t; i < NSUBJ * DIM; i += 256) {
    int s = i / DIM;
    means[i] = ws[SUMS_OFF + i] / fmaxf(nshr[s], 1.0f);
  }
  __syncthreads();

  float total = 0.0f, cntv = 0.0f;  // meaningful in thread 0 only
  for (int i = 0; i < NSUBJ; i++) {
    for (int j = i + 1; j < NSUBJ; j++) {
      const float ni = nshr[i], nj = nshr[j];
      const bool pv = (ni >= 2.0f) && (nj >= 2.0f);
      const float di = fmaxf(ni - 1.0f, 1.0f), dj = fmaxf(nj - 1.0f, 1.0f);
      const float* Gi = ws + GRAM_OFF + i * DIM * DIM;
      const float* Gj = ws + GRAM_OFF + j * DIM * DIM;
      float part = 0.0f;
      for (int e = t; e < DIM * DIM; e += 256) {
        int a = e >> 6, b = e & 63;
        // lam*I cancels in the difference (as in the reference)
        float ci = (Gi[e] - ni * means[i * DIM + a] * means[i * DIM + b]) / di;
        float cj = (Gj[e] - nj * means[j * DIM + a] * means[j * DIM + b]) / dj;
        float df = ci - cj;
        part += df * df;
      }
      red[t] = part;
      __syncthreads();
      for (int off = 128; off > 0; off >>= 1) {
        if (t < off) red[t] += red[t + off];
        __syncthreads();
      }
      if (t == 0 && pv) {
        total += sqrtf(red[0]);
        cntv += 1.0f;
      }
      __syncthreads();
    }
  }
  if (t == 0) out[0] = (cntv > 0.0f) ? (total / fmaxf(cntv, 1.0f)) : 0.0f;
}

extern "C" void kernel_launch(void* const* d_in, const int* in_sizes, int n_in,
                              void* d_out, int out_size, void* d_ws,
                              size_t ws_size, hipStream_t stream) {
  const float* emb = (const float*)d_in[0];
  const int* sid = (const int*)d_in[1];
  float* ws = (float*)d_ws;
  const int N = in_sizes[1];  // 524288 rows

  caloss_zero_ws<<<(WS_FLOATS + 255) / 256, 256, 0, stream>>>(ws);
  caloss_gram_kernel<<<64, 512, 0, stream>>>(emb, sid, ws, N);
  caloss_finalize<<<1, 256, 0, stream>>>(ws, (float*)d_out);
}